// GAEencoder_81870666596785
// MI455X (gfx1250) — compile-verified
//
#include <hip/hip_runtime.h>
#include <hip/hip_bf16.h>
#include <math.h>

typedef __attribute__((ext_vector_type(2))) float v2f;
typedef __attribute__((ext_vector_type(8))) float v8f;

// ---------------------------------------------------------------------------
// Degree / normalization
// ---------------------------------------------------------------------------
__global__ __launch_bounds__(256) void k_deg_init(float* __restrict__ deg, int n) {
    int i = blockIdx.x * 256 + threadIdx.x;
    if (i < n) deg[i] = 1.0f;  // self-loop contributes 1 to every node's degree
}

__global__ __launch_bounds__(256) void k_deg_count(const long long* __restrict__ ei,
                                                   float* __restrict__ deg,
                                                   long long E) {
    long long e = (long long)blockIdx.x * 256 + threadIdx.x;
    if (e < E) {
        long long c = ei[E + e];  // col
        atomicAdd(&deg[c], 1.0f);
    }
}

__global__ __launch_bounds__(256) void k_dinv(const float* __restrict__ deg,
                                              float* __restrict__ dinv, int n) {
    int i = blockIdx.x * 256 + threadIdx.x;
    if (i < n) dinv[i] = rsqrtf(deg[i]);  // deg >= 1 always (self-loops)
}

// ---------------------------------------------------------------------------
// GEMM1: h1[N,32] = x[N,128] @ W1[128,32]   (f32 WMMA 16x16x4)
// W1 (16 KB) staged into LDS once per block; 8 waves per block reuse it.
// A frag: lane(m=lane&15, hi=lane>>4) holds A[m][k0+2*hi .. +1]
// B frag: lane holds B[k0+2*hi .. +1][n], n = lane&15
// D:      VGPR v holds D[v + 8*hi][n]
// ---------------------------------------------------------------------------
__global__ __launch_bounds__(256) void k_gemm1(const float* __restrict__ x,
                                               const float* __restrict__ W1,
                                               float* __restrict__ h1,
                                               int nTiles) {
    __shared__ float sW1[128 * 32];

    // cooperative stage: 4096 floats = 1024 float4, 4 per thread
    {
        const float4* src = (const float4*)W1;
        float4*       dst = (float4*)sW1;
#pragma unroll
        for (int i = 0; i < 4; ++i)
            dst[threadIdx.x + 256 * i] = src[threadIdx.x + 256 * i];
    }
    __syncthreads();

    int wave = (int)((blockIdx.x * (unsigned)blockDim.x + threadIdx.x) >> 5);
    int lane = threadIdx.x & 31;
    if (wave >= nTiles) return;  // wave-uniform: EXEC stays all-ones inside

    int m  = lane & 15;
    int hi = lane >> 4;
    const float* arow = x + ((size_t)wave * 16 + m) * 128;

    v8f acc0 = {};
    v8f acc1 = {};
    for (int k0 = 0; k0 < 128; k0 += 4) {
        int kA = k0 + 2 * hi;
        v2f a = *(const v2f*)(arow + kA);  // contiguous, 8B aligned
        v2f b0, b1;
        b0.x = sW1[kA * 32 + m];
        b0.y = sW1[(kA + 1) * 32 + m];
        b1.x = sW1[kA * 32 + 16 + m];
        b1.y = sW1[(kA + 1) * 32 + 16 + m];
        acc0 = __builtin_amdgcn_wmma_f32_16x16x4_f32(false, a, false, b0,
                                                     (short)0, acc0, false, false);
        acc1 = __builtin_amdgcn_wmma_f32_16x16x4_f32(false, a, false, b1,
                                                     (short)0, acc1, false, false);
    }

    float* orow = h1 + (size_t)wave * 16 * 32;
#pragma unroll
    for (int v = 0; v < 8; ++v) {
        int M = v + 8 * hi;
        orow[(size_t)M * 32 + m]      = acc0[v];
        orow[(size_t)M * 32 + 16 + m] = acc1[v];
    }
}

// scalar fallback for (N % 16) tail rows (not used for N=100000, kept for safety)
__global__ __launch_bounds__(256) void k_gemm1_tail(const float* __restrict__ x,
                                                    const float* __restrict__ W1,
                                                    float* __restrict__ h1,
                                                    int startRow, int n) {
    int t = blockIdx.x * 256 + threadIdx.x;
    int row = startRow + (t >> 5);
    int col = t & 31;
    if (row < n) {
        float s = 0.f;
        for (int k = 0; k < 128; ++k) s += x[(size_t)row * 128 + k] * W1[(size_t)k * 32 + col];
        h1[(size_t)row * 32 + col] = s;
    }
}

// ---------------------------------------------------------------------------
// Layer-1 aggregation
// ---------------------------------------------------------------------------
// self-loop term also serves as initializer of agg1 (ws is poisoned)
__global__ __launch_bounds__(256) void k_selfloop1(const float* __restrict__ dinv,
                                                   const float* __restrict__ h1,
                                                   float* __restrict__ agg1,
                                                   long long total) {
    long long idx = (long long)blockIdx.x * 256 + threadIdx.x;
    if (idx < total) {
        long long i = idx >> 5;
        float di = dinv[i];
        agg1[idx] = di * di * h1[idx];
    }
}

// One wave per edge; lane = feature -> coalesced 128B gather + 128B atomic RMW.
// Edge index / norm / base addresses are wave-uniform: force them scalar with
// readfirstlane so the gather and atomic use the SGPR-base (saddr) form.
__global__ __launch_bounds__(256) void k_scatter1(const long long* __restrict__ ei,
                                                  const float* __restrict__ dinv,
                                                  const float* __restrict__ h1,
                                                  float* __restrict__ agg1,
                                                  long long E) {
    int lane = threadIdx.x & 31;
    long long wave = ((long long)blockIdx.x * blockDim.x + threadIdx.x) >> 5;
    if (wave >= E) return;
    // node ids < 2^31; every lane loads the same address, value is uniform
    int r = __builtin_amdgcn_readfirstlane((int)ei[wave]);
    int c = __builtin_amdgcn_readfirstlane((int)ei[E + wave]);
    float norm = dinv[r] * dinv[c];  // uniform
    float msg  = norm * h1[(size_t)r * 32 + lane];
    atomicAdd(&agg1[(size_t)c * 32 + lane], msg);
}

// ---------------------------------------------------------------------------
// GEMM2: h2[N,3] = tanh(agg1)[N,32] @ W2[32,3]   (f32 WMMA, N padded to 16)
// ---------------------------------------------------------------------------
__global__ __launch_bounds__(256) void k_gemm2(const float* __restrict__ agg1,
                                               const float* __restrict__ W2,
                                               float* __restrict__ h2,
                                               int nTiles) {
    int wave = (int)((blockIdx.x * (unsigned)blockDim.x + threadIdx.x) >> 5);
    int lane = threadIdx.x & 31;
    if (wave >= nTiles) return;

    int m  = lane & 15;
    int hi = lane >> 4;
    const float* arow = agg1 + ((size_t)wave * 16 + m) * 32;

    v8f acc = {};
    for (int k0 = 0; k0 < 32; k0 += 4) {
        int kA = k0 + 2 * hi;
        v2f a;
        a.x = tanhf(arow[kA]);
        a.y = tanhf(arow[kA + 1]);
        v2f b;
        b.x = (m < 3) ? W2[(size_t)kA * 3 + m] : 0.f;       // pad cols 3..15 with 0
        b.y = (m < 3) ? W2[(size_t)(kA + 1) * 3 + m] : 0.f;
        acc = __builtin_amdgcn_wmma_f32_16x16x4_f32(false, a, false, b,
                                                    (short)0, acc, false, false);
    }

    if (m < 3) {  // divergence only after all WMMAs retired
#pragma unroll
        for (int v = 0; v < 8; ++v) {
            int M = v + 8 * hi;
            h2[((size_t)wave * 16 + M) * 3 + m] = acc[v];
        }
    }
}

__global__ __launch_bounds__(256) void k_gemm2_tail(const float* __restrict__ agg1,
                                                    const float* __restrict__ W2,
                                                    float* __restrict__ h2,
                                                    int startRow, int n) {
    int t = blockIdx.x * 256 + threadIdx.x;
    int row = startRow + t / 3;
    int col = t % 3;
    if (row < n) {
        float s = 0.f;
        for (int k = 0; k < 32; ++k)
            s += tanhf(agg1[(size_t)row * 32 + k]) * W2[(size_t)k * 3 + col];
        h2[(size_t)row * 3 + col] = s;
    }
}

// ---------------------------------------------------------------------------
// Layer-2 aggregation (into d_out)
// ---------------------------------------------------------------------------
__global__ __launch_bounds__(256) void k_zinit(const float* __restrict__ dinv,
                                               const float* __restrict__ h2,
                                               float* __restrict__ z, int total) {
    int idx = blockIdx.x * 256 + threadIdx.x;
    if (idx < total) {
        int i = idx / 3;
        float di = dinv[i];
        z[idx] = di * di * h2[idx];
    }
}

__global__ __launch_bounds__(256) void k_scatter2(const long long* __restrict__ ei,
                                                  const float* __restrict__ dinv,
                                                  const float* __restrict__ h2,
                                                  float* __restrict__ z,
                                                  long long E) {
    long long e = (long long)blockIdx.x * 256 + threadIdx.x;
    if (e >= E) return;
    long long r = ei[e];
    long long c = ei[E + e];
    float norm = dinv[r] * dinv[c];
    atomicAdd(&z[c * 3 + 0], norm * h2[r * 3 + 0]);
    atomicAdd(&z[c * 3 + 1], norm * h2[r * 3 + 1]);
    atomicAdd(&z[c * 3 + 2], norm * h2[r * 3 + 2]);
}

// ---------------------------------------------------------------------------
extern "C" void kernel_launch(void* const* d_in, const int* in_sizes, int n_in,
                              void* d_out, int out_size, void* d_ws, size_t ws_size,
                              hipStream_t stream) {
    const float*     x  = (const float*)d_in[0];      // [N,128] f32
    const long long* ei = (const long long*)d_in[1];  // [2,E] int64
    const float*     W1 = (const float*)d_in[2];      // [128,32] f32
    const float*     W2 = (const float*)d_in[3];      // [32,3] f32
    float*           z  = (float*)d_out;              // [N,3] f32

    const int       N = in_sizes[0] / 128;
    const long long E = (long long)in_sizes[1] / 2;

    // workspace layout (floats): deg[N] | dinv[N] | h1[N*32] | agg1[N*32] | h2[N*3]
    float* ws   = (float*)d_ws;
    float* deg  = ws;
    float* dinv = deg + N;
    float* h1   = dinv + N;
    float* agg1 = h1 + (size_t)N * 32;
    float* h2   = agg1 + (size_t)N * 32;

    const dim3 blk(256);
    const unsigned gN   = (unsigned)((N + 255) / 256);
    const unsigned gE   = (unsigned)((E + 255) / 256);
    const unsigned gEw  = (unsigned)((E * 32 + 255) / 256);  // wave-per-edge
    const unsigned gN32 = (unsigned)(((long long)N * 32 + 255) / 256);
    const unsigned gN3  = (unsigned)((N * 3 + 255) / 256);

    // normalization
    k_deg_init <<<gN, blk, 0, stream>>>(deg, N);
    k_deg_count<<<gE, blk, 0, stream>>>(ei, deg, E);
    k_dinv     <<<gN, blk, 0, stream>>>(deg, dinv, N);

    // layer 1
    const int tiles = N / 16;
    const int rem   = N - tiles * 16;
    if (tiles) k_gemm1<<<(unsigned)((tiles + 7) / 8), blk, 0, stream>>>(x, W1, h1, tiles);
    if (rem)   k_gemm1_tail<<<(unsigned)((rem * 32 + 255) / 256), blk, 0, stream>>>(x, W1, h1, tiles * 16, N);
    k_selfloop1<<<gN32, blk, 0, stream>>>(dinv, h1, agg1, (long long)N * 32);
    k_scatter1 <<<gEw, blk, 0, stream>>>(ei, dinv, h1, agg1, E);

    // layer 2
    if (tiles) k_gemm2<<<(unsigned)((tiles + 7) / 8), blk, 0, stream>>>(agg1, W2, h2, tiles);
    if (rem)   k_gemm2_tail<<<(unsigned)((rem * 3 + 255) / 256), blk, 0, stream>>>(agg1, W2, h2, tiles * 16, N);
    k_zinit    <<<gN3, blk, 0, stream>>>(dinv, h2, z, N * 3);
    k_scatter2 <<<gE, blk, 0, stream>>>(ei, dinv, h2, z, E);
}